// RoIBBox_20057497272710
// MI455X (gfx1250) — compile-verified
//
#include <hip/hip_runtime.h>
#include <hip/hip_bf16.h>
#include <math.h>

// ---------------- problem constants (match reference) ----------------
#define PRE_NMS   2000
#define PRE_PAD   2048          // power of two for bitonic sort
#define POST_NMS  300
#define NMS_T     0.7f
#define EPS_F     1e-8f
#define HDIM      100
#define WDIM      100
#define ADIM      9
#define NANCH     (HDIM * WDIM * ADIM)   // 90000
#define THREADS   1024
#define NWAVES    (THREADS / 32)
#define EQ_CAP    256

typedef __attribute__((ext_vector_type(2))) float v2f;
typedef __attribute__((ext_vector_type(8))) float v8f;

__device__ __forceinline__ unsigned wave_ballot32(bool p) {
#if __has_builtin(__builtin_amdgcn_ballot_w32)
    return __builtin_amdgcn_ballot_w32(p);
#else
    return (unsigned)__ballot((int)p);   // wave32: low 32 bits valid
#endif
}

// score (uniform [0,1), non-negative float) -> order-preserving u32 key
__device__ __forceinline__ unsigned score_key(float s) {
    return __float_as_uint(s);
}

__global__ __launch_bounds__(THREADS)
void roi_bbox_kernel(const float* __restrict__ deltas,   // [B, N, 4]
                     const float* __restrict__ labels,   // [B, N]
                     const float* __restrict__ anchors,  // [N, 4]
                     float* __restrict__ out,            // [B, POST_NMS, 4]
                     int B) {
    const int b   = blockIdx.x;
    const int tid = threadIdx.x;
    const int lane = tid & 31;
    const int wave = tid >> 5;
    if (b >= B) return;

    __shared__ unsigned           s_hist[256];
    __shared__ unsigned long long s_cand[PRE_PAD];   // (key<<32) | ~index
    __shared__ float              s_box[PRE_PAD][4];
    __shared__ float              s_area[PRE_PAD];
    __shared__ unsigned char      s_keep[PRE_PAD];
    __shared__ unsigned           s_eq[EQ_CAP];
    __shared__ unsigned           s_prefix;
    __shared__ unsigned           s_remaining;
    __shared__ int                s_cntGT, s_cntEQ;

    const float* __restrict__ lab_b = labels + (size_t)b * NANCH;

    // ---------------- phase 1: radix-select top-PRE_NMS threshold ----------------
    if (tid == 0) { s_prefix = 0u; s_remaining = PRE_NMS; }
    __syncthreads();

    for (int d = 0; d < 4; ++d) {
        for (int i = tid; i < 256; i += THREADS) s_hist[i] = 0u;
        __syncthreads();
        const unsigned pref = s_prefix;
        const int sh_match = 32 - 8 * d;
        const int sh_dig   = 24 - 8 * d;
        for (int n = tid; n < NANCH; n += THREADS) {
            __builtin_prefetch(lab_b + n + 4 * THREADS, 0, 1);  // global_prefetch_b8
            unsigned k = score_key(lab_b[n]);
            bool match = (d == 0) || ((k >> sh_match) == pref);
            if (match) atomicAdd(&s_hist[(k >> sh_dig) & 255u], 1u);
        }
        __syncthreads();
        if (tid == 0) {
            unsigned rem = s_remaining, acc = 0;
            int v = 255;
            for (; v > 0; --v) {
                unsigned c = s_hist[v];
                if (acc + c >= rem) break;
                acc += c;
            }
            s_remaining = rem - acc;
            s_prefix = (s_prefix << 8) | (unsigned)v;
        }
        __syncthreads();
    }

    const unsigned Tkey = s_prefix;   // exact 32-bit threshold key

    // ---------------- phase 2: compact candidates into LDS ----------------
    for (int i = tid; i < PRE_PAD; i += THREADS) s_cand[i] = 0ull;
    if (tid == 0) { s_cntGT = 0; s_cntEQ = 0; }
    __syncthreads();

    for (int n = tid; n < NANCH; n += THREADS) {
        unsigned k = score_key(lab_b[n]);
        if (k > Tkey) {
            int p = atomicAdd(&s_cntGT, 1);
            if (p < PRE_NMS)
                s_cand[p] = ((unsigned long long)k << 32) | (unsigned)(~(unsigned)n);
        } else if (k == Tkey) {
            int p = atomicAdd(&s_cntEQ, 1);
            if (p < EQ_CAP) s_eq[p] = (unsigned)n;
        }
    }
    __syncthreads();

    if (tid == 0) {
        int g  = s_cntGT; if (g > PRE_NMS) g = PRE_NMS;
        int ne = PRE_NMS - g;                       // needed from == set
        int avail = s_cntEQ; if (avail > EQ_CAP) avail = EQ_CAP;
        // ties broken by smallest index (matches lax.top_k): insertion sort asc
        for (int i = 1; i < avail; ++i) {
            unsigned x = s_eq[i]; int j = i - 1;
            while (j >= 0 && s_eq[j] > x) { s_eq[j + 1] = s_eq[j]; --j; }
            s_eq[j + 1] = x;
        }
        int take = (ne < avail) ? ne : avail;
        for (int i = 0; i < take && (g + i) < PRE_NMS; ++i)
            s_cand[g + i] = ((unsigned long long)Tkey << 32) | (unsigned)(~s_eq[i]);
    }
    __syncthreads();

    // ---------------- phase 3: bitonic sort 2048, descending (key desc, idx asc) ----------------
    for (unsigned k = 2; k <= PRE_PAD; k <<= 1) {
        for (unsigned j = k >> 1; j > 0; j >>= 1) {
            for (unsigned i = tid; i < PRE_PAD; i += THREADS) {
                unsigned ixj = i ^ j;
                if (ixj > i) {
                    unsigned long long a = s_cand[i], c = s_cand[ixj];
                    bool sw = ((i & k) == 0) ? (a < c) : (a > c);
                    if (sw) { s_cand[i] = c; s_cand[ixj] = a; }
                }
            }
            __syncthreads();
        }
    }

    // ---------------- phase 4: decode only the selected 2000 boxes ----------------
    for (int t = tid; t < PRE_PAD; t += THREADS) {
        float b0 = 0.f, b1 = 0.f, b2 = 0.f, b3 = 0.f, ar = 0.f;
        unsigned char kp = 0;
        unsigned long long e = s_cand[t];
        if (t < PRE_NMS && e != 0ull) {
            unsigned n = ~(unsigned)(e & 0xFFFFFFFFull);
            if (n < NANCH) {
                const float* ap = anchors + (size_t)n * 4;
                float ay1 = ap[0], ax1 = ap[1], ay2 = ap[2], ax2 = ap[3];
                const float* dp = deltas + ((size_t)b * NANCH + n) * 4;
                float dy = dp[0], dx = dp[1], dh = dp[2], dw = dp[3];
                float ah = ay2 - ay1, aw = ax2 - ax1;
                float acy = ay1 + 0.5f * ah, acx = ax1 + 0.5f * aw;
                float h = __expf(dh) * ah, w = __expf(dw) * aw;
                float cy = dy * ah + acy, cx = dx * aw + acx;
                b0 = cy - 0.5f * h; b1 = cx - 0.5f * w;
                b2 = b0 + h;        b3 = b1 + w;
                ar = fmaxf(b2 - b0, 0.f) * fmaxf(b3 - b1, 0.f);
                kp = 1;
            }
        }
        s_box[t][0] = b0; s_box[t][1] = b1; s_box[t][2] = b2; s_box[t][3] = b3;
        s_area[t] = ar; s_keep[t] = kp;
    }
    __syncthreads();

    // ---------------- phase 5: exact tiled greedy NMS ----------------
    // 16-row tiles processed in score order. Intra-tile: wave 0, ballot-serial.
    // Tile->later-columns sweep: all waves, 16x16 tiles; areaSum[i][j] =
    // area_i + area_j via rank-2 outer product on v_wmma_f32_16x16x4_f32.
    for (int tb = 0; tb < PRE_NMS; tb += 16) {
        if (wave == 0) {
            const int r = lane & 15;
            const bool act = lane < 16;
            float y1 = 0.f, x1 = 0.f, y2 = 0.f, x2 = 0.f, ar = 0.f;
            if (act) {
                y1 = s_box[tb + r][0]; x1 = s_box[tb + r][1];
                y2 = s_box[tb + r][2]; x2 = s_box[tb + r][3];
                ar = s_area[tb + r];
            }
            unsigned mask = wave_ballot32(act && (s_keep[tb + r] != 0)) & 0xFFFFu;
            for (int i = 0; i < 16; ++i) {
                if (!((mask >> i) & 1u)) continue;   // uniform (ballot in SGPR)
                float iy1 = s_box[tb + i][0], ix1 = s_box[tb + i][1];
                float iy2 = s_box[tb + i][2], ix2 = s_box[tb + i][3];
                float iar = s_area[tb + i];
                float yy1 = fmaxf(iy1, y1), xx1 = fmaxf(ix1, x1);
                float yy2 = fminf(iy2, y2), xx2 = fminf(ix2, x2);
                float inter = fmaxf(yy2 - yy1, 0.f) * fmaxf(xx2 - xx1, 0.f);
                bool sup = act && (r > i) &&
                           (inter * (1.f + NMS_T) > NMS_T * (iar + ar + EPS_F));
                mask &= ~(wave_ballot32(sup) & 0xFFFFu);
            }
            if (act && !((mask >> r) & 1u)) {
                s_keep[tb + r] = 0;
                s_box[tb + r][0] = 0.f; s_box[tb + r][1] = 0.f;
                s_box[tb + r][2] = 0.f; s_box[tb + r][3] = 0.f;
                s_area[tb + r] = 0.f;   // zero area => can never suppress
            }
        }
        __syncthreads();

        // parallel sweep: each wave owns 16-column blocks
        for (int cb = tb + 16 + wave * 16; cb < PRE_PAD; cb += NWAVES * 16) {
            // Build rank-2 operands for areaSum = area_row + area_col.
            // A (16x4 f32): lanes 0-15 hold K=0 (v.x) and K=1 (v.y); lanes 16-31 hold K=2,3 (zero).
            // B (4x16 f32): lanes 0-15 hold K=0 (v.x) and K=1 (v.y); lanes 16-31 hold K=2,3 (zero).
            v2f av, bv;
            if (lane < 16) {
                av.x = s_area[tb + lane]; av.y = 1.0f;
                bv.x = 1.0f;              bv.y = s_area[cb + lane];
            } else {
                av.x = 0.f; av.y = 0.f; bv.x = 0.f; bv.y = 0.f;
            }
            v8f acc = {0.f, 0.f, 0.f, 0.f, 0.f, 0.f, 0.f, 0.f};
#if __has_builtin(__builtin_amdgcn_wmma_f32_16x16x4_f32)
            acc = __builtin_amdgcn_wmma_f32_16x16x4_f32(
                false, av, false, bv, (short)0, acc, false, false);
#else
            {   // fallback: explicit area sums (no wmma available)
                const int Nc0 = lane & 15;
                const int Mb0 = (lane < 16) ? 0 : 8;
                #pragma unroll
                for (int r = 0; r < 8; ++r)
                    acc[r] = s_area[tb + Mb0 + r] + s_area[cb + Nc0];
            }
#endif
            // C/D layout: VGPR r, lanes 0-15 -> M=r, lanes 16-31 -> M=r+8; N = lane%16
            const int Nc = lane & 15;
            const int Mbase = (lane < 16) ? 0 : 8;
            const float cy1 = s_box[cb + Nc][0], cx1 = s_box[cb + Nc][1];
            const float cy2 = s_box[cb + Nc][2], cx2 = s_box[cb + Nc][3];
            bool kill = false;
            #pragma unroll
            for (int r = 0; r < 8; ++r) {
                const int M = tb + Mbase + r;
                float ry1 = s_box[M][0], rx1 = s_box[M][1];
                float ry2 = s_box[M][2], rx2 = s_box[M][3];
                float yy1 = fmaxf(ry1, cy1), xx1 = fmaxf(rx1, cx1);
                float yy2 = fminf(ry2, cy2), xx2 = fminf(rx2, cx2);
                float inter = fmaxf(yy2 - yy1, 0.f) * fmaxf(xx2 - xx1, 0.f);
                // iou > t  <=>  inter*(1+t) > t*(area_i+area_j+eps)
                kill |= (inter * (1.f + NMS_T) > NMS_T * (acc[r] + EPS_F));
            }
            if (kill) s_keep[cb + Nc] = 0;
        }
        __syncthreads();
    }

    // ---------------- phase 6: compact kept boxes (score order), clip, pad ----------------
    float* ob = out + (size_t)b * POST_NMS * 4;
    for (int i = tid; i < POST_NMS * 4; i += THREADS) ob[i] = 0.f;
    __threadfence();
    __syncthreads();
    if (tid == 0) {
        int cnt = 0;
        for (int i = 0; i < PRE_NMS && cnt < POST_NMS; ++i) {
            if (s_keep[i]) {
                ob[cnt * 4 + 0] = fminf(fmaxf(s_box[i][0], 0.f), 1.f);
                ob[cnt * 4 + 1] = fminf(fmaxf(s_box[i][1], 0.f), 1.f);
                ob[cnt * 4 + 2] = fminf(fmaxf(s_box[i][2], 0.f), 1.f);
                ob[cnt * 4 + 3] = fminf(fmaxf(s_box[i][3], 0.f), 1.f);
                ++cnt;
            }
        }
    }
}

extern "C" void kernel_launch(void* const* d_in, const int* in_sizes, int n_in,
                              void* d_out, int out_size, void* d_ws, size_t ws_size,
                              hipStream_t stream) {
    (void)n_in; (void)d_ws; (void)ws_size; (void)out_size;
    const float* deltas  = (const float*)d_in[0];  // [B, H, W, A*4]
    const float* labels  = (const float*)d_in[1];  // [B, H, W, A]
    const float* anchors = (const float*)d_in[2];  // [N, 4]
    float* out = (float*)d_out;                    // [B, POST_NMS, 4]
    int B = in_sizes[1] / NANCH;
    if (B <= 0) B = 1;
    roi_bbox_kernel<<<dim3(B), dim3(THREADS), 0, stream>>>(deltas, labels, anchors, out, B);
}